// LGC_45191645889206
// MI455X (gfx1250) — compile-verified
//
#include <hip/hip_runtime.h>

typedef __attribute__((ext_vector_type(2))) float v2f;
typedef __attribute__((ext_vector_type(8))) float v8f;

// D = A(16x4) * B(4x16) + C, fp32, wave32
__device__ __forceinline__ v8f wmma4(v2f a, v2f b, v8f c) {
  return __builtin_amdgcn_wmma_f32_16x16x4_f32(
      /*neg_a=*/false, a, /*neg_b=*/false, b,
      /*c_mod=*/(short)0, c, /*reuse_a=*/false, /*reuse_b=*/false);
}

__device__ __forceinline__ v8f v8zero() {
  v8f z = {0.f, 0.f, 0.f, 0.f, 0.f, 0.f, 0.f, 0.f};
  return z;
}

// Shared tail: bufIn[16x256] -> (W2[256x256]+B2, relu) -> bufOut[16x256]
//              bufOut -> (W3[256x128]+B3) -> gout[row0.., 128]
__device__ __forceinline__ void mlp_tail_256_256_128(
    const float* __restrict__ W2, const float* __restrict__ B2,
    const float* __restrict__ W3, const float* __restrict__ B3,
    const float* bufIn, float* bufOut,
    float* __restrict__ gout, int row0, int nrows,
    int m, int hi, int o) {
  // ---- Layer 2: [16x256] @ [256x256], relu ----
  for (int ntb = 0; ntb < 2; ++ntb) {
    v8f acc[8];
#pragma unroll
    for (int nt = 0; nt < 8; ++nt) acc[nt] = v8zero();
    for (int kk = 0; kk < 64; ++kk) {
      const int kb = kk * 4;
      v2f a = *(const v2f*)(bufIn + m * 256 + kb + o);
#pragma unroll
      for (int nt = 0; nt < 8; ++nt) {
        const int n = ntb * 128 + nt * 16 + m;
        v2f b;
        b.x = W2[(kb + o) * 256 + n];
        b.y = W2[(kb + o + 1) * 256 + n];
        acc[nt] = wmma4(a, b, acc[nt]);
      }
    }
#pragma unroll
    for (int nt = 0; nt < 8; ++nt) {
      const int n = ntb * 128 + nt * 16 + m;
      const float bb = B2[n];
#pragma unroll
      for (int g = 0; g < 8; ++g) {
        float v = acc[nt][g] + bb;
        bufOut[(g + 8 * hi) * 256 + n] = v > 0.f ? v : 0.f;
      }
    }
  }
  __syncthreads();
  // ---- Layer 3: [16x256] @ [256x128], +bias, store to global ----
  {
    v8f acc[8];
#pragma unroll
    for (int nt = 0; nt < 8; ++nt) acc[nt] = v8zero();
    for (int kk = 0; kk < 64; ++kk) {
      const int kb = kk * 4;
      v2f a = *(const v2f*)(bufOut + m * 256 + kb + o);
#pragma unroll
      for (int nt = 0; nt < 8; ++nt) {
        const int n = nt * 16 + m;
        v2f b;
        b.x = W3[(kb + o) * 128 + n];
        b.y = W3[(kb + o + 1) * 128 + n];
        acc[nt] = wmma4(a, b, acc[nt]);
      }
    }
#pragma unroll
    for (int nt = 0; nt < 8; ++nt) {
      const int n = nt * 16 + m;
      const float bb = B3[n];
#pragma unroll
      for (int g = 0; g < 8; ++g) {
        const int r = row0 + g + 8 * hi;
        if (r < nrows) gout[(size_t)r * 128 + n] = acc[nt][g] + bb;
      }
    }
  }
}

// Kernel 1: y = MLP1(x), 128 -> 256 -> 256 -> 128, one wave per 16 rows
__global__ __launch_bounds__(32) void node_mlp1_kernel(
    const float* __restrict__ x,
    const float* __restrict__ W1, const float* __restrict__ B1,
    const float* __restrict__ W2, const float* __restrict__ B2,
    const float* __restrict__ W3, const float* __restrict__ B3,
    float* __restrict__ y, int nrows) {
  __shared__ float buf[2][16 * 256];
  const int lane = threadIdx.x & 31;
  const int m = lane & 15;
  const int hi = lane >> 4;
  const int o = hi * 2;
  const int row0 = blockIdx.x * 16;

  int rm = row0 + m;
  if (rm >= nrows) rm = nrows - 1;
  const float* xrow = x + (size_t)rm * 128;

  // ---- Layer 1: [16x128] @ W1[128x256], relu -> buf[0] ----
  for (int ntb = 0; ntb < 2; ++ntb) {
    v8f acc[8];
#pragma unroll
    for (int nt = 0; nt < 8; ++nt) acc[nt] = v8zero();
    for (int kk = 0; kk < 32; ++kk) {
      const int kb = kk * 4;
      v2f a = *(const v2f*)(xrow + kb + o);
#pragma unroll
      for (int nt = 0; nt < 8; ++nt) {
        const int n = ntb * 128 + nt * 16 + m;
        v2f b;
        b.x = W1[(kb + o) * 256 + n];
        b.y = W1[(kb + o + 1) * 256 + n];
        acc[nt] = wmma4(a, b, acc[nt]);
      }
    }
#pragma unroll
    for (int nt = 0; nt < 8; ++nt) {
      const int n = ntb * 128 + nt * 16 + m;
      const float bb = B1[n];
#pragma unroll
      for (int g = 0; g < 8; ++g) {
        float v = acc[nt][g] + bb;
        buf[0][(g + 8 * hi) * 256 + n] = v > 0.f ? v : 0.f;
      }
    }
  }
  __syncthreads();
  mlp_tail_256_256_128(W2, B2, W3, B3, buf[0], buf[1], y, row0, nrows, m, hi, o);
}

// Kernel 2: out = MLP2(concat(y[row[8n+c]] c=0..7, x[n])), 1152 -> 256 -> 256 -> 128
__global__ __launch_bounds__(32) void node_mlp2_kernel(
    const float* __restrict__ x,
    const float* __restrict__ y,
    const int* __restrict__ erow,  // edge_index[0,:], E = 8*nrows entries
    const float* __restrict__ W1, const float* __restrict__ B1,
    const float* __restrict__ W2, const float* __restrict__ B2,
    const float* __restrict__ W3, const float* __restrict__ B3,
    float* __restrict__ out, int nrows) {
  __shared__ float buf[2][16 * 256];
  const int lane = threadIdx.x & 31;
  const int m = lane & 15;
  const int hi = lane >> 4;
  const int o = hi * 2;
  const int row0 = blockIdx.x * 16;

  int node = row0 + m;
  if (node >= nrows) node = nrows - 1;

  const float* srcs[9];
#pragma unroll
  for (int c = 0; c < 8; ++c) {
    srcs[c] = y + (size_t)erow[node * 8 + c] * 128;
  }
  srcs[8] = x + (size_t)node * 128;

  // ---- Layer 1: gathered [16x1152] @ W1[1152x256], relu -> buf[0] ----
  for (int ntb = 0; ntb < 2; ++ntb) {
    v8f acc[8];
#pragma unroll
    for (int nt = 0; nt < 8; ++nt) acc[nt] = v8zero();
#pragma unroll
    for (int c = 0; c < 9; ++c) {
      const float* s = srcs[c];
      const int kg0 = c * 128;
      for (int kk = 0; kk < 32; ++kk) {
        const int kb = kk * 4;
        v2f a = *(const v2f*)(s + kb + o);
        const int kg = kg0 + kb + o;
#pragma unroll
        for (int nt = 0; nt < 8; ++nt) {
          const int n = ntb * 128 + nt * 16 + m;
          v2f b;
          b.x = W1[kg * 256 + n];
          b.y = W1[(kg + 1) * 256 + n];
          acc[nt] = wmma4(a, b, acc[nt]);
        }
      }
    }
#pragma unroll
    for (int nt = 0; nt < 8; ++nt) {
      const int n = ntb * 128 + nt * 16 + m;
      const float bb = B1[n];
#pragma unroll
      for (int g = 0; g < 8; ++g) {
        float v = acc[nt][g] + bb;
        buf[0][(g + 8 * hi) * 256 + n] = v > 0.f ? v : 0.f;
      }
    }
  }
  __syncthreads();
  mlp_tail_256_256_128(W2, B2, W3, B3, buf[0], buf[1], out, row0, nrows, m, hi, o);
}

extern "C" void kernel_launch(void* const* d_in, const int* in_sizes, int n_in,
                              void* d_out, int out_size, void* d_ws, size_t ws_size,
                              hipStream_t stream) {
  // setup_inputs order:
  // 0:x 1:edge_index 2:edge_slot 3:edge_attr 4:u 5:batch
  // 6..11: mlp1 (W 128x256, b 256, W 256x256, b 256, W 256x128, b 128)
  // 12..17: mlp2 (W 1152x256, b 256, W 256x256, b 256, W 256x128, b 128)
  const float* x = (const float*)d_in[0];
  const int* edge_index = (const int*)d_in[1];  // [2,E]; first E = row
  const float* W1a = (const float*)d_in[6];
  const float* b1a = (const float*)d_in[7];
  const float* W1b = (const float*)d_in[8];
  const float* b1b = (const float*)d_in[9];
  const float* W1c = (const float*)d_in[10];
  const float* b1c = (const float*)d_in[11];
  const float* W2a = (const float*)d_in[12];
  const float* b2a = (const float*)d_in[13];
  const float* W2b = (const float*)d_in[14];
  const float* b2b = (const float*)d_in[15];
  const float* W2c = (const float*)d_in[16];
  const float* b2c = (const float*)d_in[17];

  const int nrows = in_sizes[0] / 128;  // N
  float* y = (float*)d_ws;              // N x 128 intermediate (51.2 MB)
  float* out = (float*)d_out;

  const int tiles = (nrows + 15) / 16;
  node_mlp1_kernel<<<dim3(tiles), dim3(32), 0, stream>>>(
      x, W1a, b1a, W1b, b1b, W1c, b1c, y, nrows);
  node_mlp2_kernel<<<dim3(tiles), dim3(32), 0, stream>>>(
      x, y, edge_index, W2a, b2a, W2b, b2b, W2c, b2c, out, nrows);
}